// Supervisor_21818433863921
// MI455X (gfx1250) — compile-verified
//
#include <hip/hip_runtime.h>
#include <math.h>

typedef __attribute__((ext_vector_type(2))) float v2f;
typedef __attribute__((ext_vector_type(8))) float v8f;

#define NB      64
#define TT      32
#define M_ROWS  2048            // NB*TT
#define KDIM    12288           // 3*64*64
#define KSPLIT  8
#define KCHUNK  (KDIM / KSPLIT) // 1536
#define MTILES  (M_ROWS / 16)   // 128
#define EPS_FIX 0.095f

// ---------------------------------------------------------------------------
// K0: pack the 12 useful columns of W_enc (cols 8o..8o+3) into Bt[16][KDIM],
// transposed so B-lane loads in the GEMM are contiguous 8-byte reads.
// Rows 12..15 are zero padding (N padded to 16 for the WMMA tile).
// ---------------------------------------------------------------------------
__global__ void prep_b_kernel(const float* __restrict__ W, float* __restrict__ Bt) {
    int id = blockIdx.x * blockDim.x + threadIdx.x;   // 16*KDIM threads
    int c  = id / KDIM;
    int k  = id - c * KDIM;
    float v = 0.0f;
    if (c < 12) {
        int wc = (c >> 2) * 8 + (c & 3);              // 0-3, 8-11, 16-19
        v = W[(size_t)k * 24 + wc];
    }
    Bt[id] = v;
}

// ---------------------------------------------------------------------------
// K1: fp32 WMMA GEMM.  grid = (128 M-tiles, 8 K-splits), one wave32 each.
// A (16x4 fp32) layout: lane L holds M=L&15; VGPR pair covers K = 2*(L>>4)+{0,1}.
// B (4x16 fp32) layout: lane L holds N=L&15; VGPR pair covers K = 2*(L>>4)+{0,1}.
// C/D: 8 VGPRs, lane half selects M+8.
// ---------------------------------------------------------------------------
__global__ void gemm_wmma_kernel(const float* __restrict__ x,
                                 const float* __restrict__ Bt,
                                 float* __restrict__ part) {
    const int tile = blockIdx.x;
    const int ks   = blockIdx.y;
    const int lane = threadIdx.x;        // 0..31, full wave, EXEC all ones
    const int rc   = lane & 15;          // M row for A, N col for B
    const int half = lane >> 4;          // K sub-offset (0 or 2)

    const float* ap = x  + (size_t)(tile * 16 + rc) * KDIM + ks * KCHUNK + half * 2;
    const float* bp = Bt + (size_t)rc * KDIM               + ks * KCHUNK + half * 2;

    v8f acc = {0.f, 0.f, 0.f, 0.f, 0.f, 0.f, 0.f, 0.f};

#pragma unroll 8
    for (int kk = 0; kk < KCHUNK; kk += 4) {
        v2f a  = *(const v2f*)(ap + kk);
        v2f bb = *(const v2f*)(bp + kk);
        // 8 args: (neg_a, A, neg_b, B, c_mod, C, reuse_a, reuse_b)
        acc = __builtin_amdgcn_wmma_f32_16x16x4_f32(
                false, a, false, bb, (short)0, acc, false, false);
    }

    float* op = part + ((size_t)(tile * KSPLIT + ks)) * 256 + lane * 8;
#pragma unroll
    for (int i = 0; i < 8; ++i) op[i] = acc[i];
}

// ---------------------------------------------------------------------------
// K2: reduce the 8 K-split partials, add bias, apply sigmoid constraints.
// z element (m, o, cc):  tile=m>>4, r=m&15 -> lane=((r>>3)<<4)|(o*4+cc), vgpr=r&7
// ---------------------------------------------------------------------------
__global__ void reduce_constrain_kernel(const float* __restrict__ part,
                                        const float* __restrict__ b_enc,
                                        float* __restrict__ z) {
    int t   = blockIdx.x * blockDim.x + threadIdx.x;  // < 2048*12
    int m   = t / 12;
    int col = t - m * 12;        // o*4+cc, also the N index in the tile
    int o   = col >> 2;
    int cc  = col & 3;
    int tile = m >> 4;
    int r    = m & 15;
    int lane = ((r >> 3) << 4) | col;
    int i    = r & 7;

    float s = 0.0f;
#pragma unroll
    for (int ks = 0; ks < KSPLIT; ++ks)
        s += part[(size_t)(tile * KSPLIT + ks) * 256 + lane * 8 + i];
    s += b_enc[o * 8 + cc];

    float sg = 1.0f / (1.0f + expf(-s));
    float val;
    if (cc == 0)      val = sg * 0.7f + 0.1f;              // [MIN,MAX]_OBJ_SCALE
    else if (cc == 1) val = sg * 0.5f + 0.75f;             // [MIN,MAX]_Y_SCALE
    else              val = (2.0f * sg - 1.0f) * 0.9f;     // OBJ_POS_BOUND
    z[t] = val;
}

// ---------------------------------------------------------------------------
// K3: per-batch sequential matching scan + _fix_supair + output transform.
// One 32-thread block per batch element; z lives in LDS.
// ---------------------------------------------------------------------------
__global__ void match_fix_out_kernel(const float* __restrict__ z,
                                     float* __restrict__ out) {
    __shared__ float Z [TT * 12];
    __shared__ float ZF[TT * 12];
    const int b    = blockIdx.x;
    const int lane = threadIdx.x;

    for (int e = lane; e < TT * 12; e += 32)
        Z[e] = z[(size_t)b * TT * 12 + e];
    __syncthreads();

    if (lane == 0) {
        for (int t = 1; t < TT; ++t) {
            float* prev = &Z[(t - 1) * 12];
            float* cur  = &Z[t * 12];
            float e3[3][3];
            for (int i = 0; i < 3; ++i)
                for (int j = 0; j < 3; ++j) {
                    float dx = prev[i * 4 + 2] - cur[j * 4 + 2];
                    float dy = prev[i * 4 + 3] - cur[j * 4 + 3];
                    e3[i][j] = dx * dx + dy * dy;
                }
            int idx[3];
            for (int i = 0; i < 3; ++i) {            // jnp.argmin: first min
                int bj = 0; float bv = e3[i][0];
                for (int j = 1; j < 3; ++j)
                    if (e3[i][j] < bv) { bv = e3[i][j]; bj = j; }
                idx[i] = bj;
            }
            bool ok = (idx[1] != idx[0]) && (idx[2] != idx[1]) && (idx[0] != idx[2]);
            if (!ok) {                               // greedy assignment
                float used[3] = {0.f, 0.f, 0.f};
                for (int i = 0; i < 3; ++i) {
                    int bj = 0; float bv = e3[i][0] + used[0] * 1e12f;
                    for (int j = 1; j < 3; ++j) {
                        float v = e3[i][j] + used[j] * 1e12f;
                        if (v < bv) { bv = v; bj = j; }
                    }
                    used[bj] = 1.f;
                    idx[i] = bj;
                }
            }
            float tmp[12];
            for (int i = 0; i < 3; ++i)
                for (int c = 0; c < 4; ++c)
                    tmp[i * 4 + c] = cur[idx[i] * 4 + c];
            for (int e = 0; e < 12; ++e) cur[e] = tmp[e];
        }
    }
    __syncthreads();

    // _fix_supair: mask from scale-column diffs, smooth from matched z
    for (int e = lane; e < TT * 12; e += 32) {
        int t = e / 12, rem = e - t * 12, o = rem >> 2, c = rem & 3;
        int cs = c & 1;                              // mask column = c % 2
        float zc     = Z[t * 12 + o * 4 + cs];
        float prevd  = (t == 0)      ? 0.f : fabsf(zc - Z[(t - 1) * 12 + o * 4 + cs]);
        float afterd = (t == TT - 1) ? 0.f : fabsf(Z[(t + 1) * 12 + o * 4 + cs] - zc);
        bool  msk    = (prevd > EPS_FIX) && (afterd > EPS_FIX);
        float sm     = (t == 0 || t == TT - 1) ? 0.f
                     : 0.5f * (Z[(t - 1) * 12 + rem] + Z[(t + 1) * 12 + rem]);
        ZF[e] = msk ? sm : Z[e];
    }
    __syncthreads();

    // output: out[b,t,o,0:2] = pos(t+1)+1 ; out[b,t,o,2:4] = 10*(pos(t+1)-pos(t))
    for (int e = lane; e < (TT - 1) * 12; e += 32) {
        int t = e / 12, rem = e - t * 12, o = rem >> 2, c = rem & 3;
        float val;
        if (c < 2)
            val = ZF[(t + 1) * 12 + o * 4 + c + 2] + 1.0f;
        else
            val = 10.0f * (ZF[(t + 1) * 12 + o * 4 + c] - ZF[t * 12 + o * 4 + c]);
        out[((size_t)b * (TT - 1) + t) * 12 + rem] = val;
    }
}

// ---------------------------------------------------------------------------
extern "C" void kernel_launch(void* const* d_in, const int* in_sizes, int n_in,
                              void* d_out, int out_size, void* d_ws, size_t ws_size,
                              hipStream_t stream) {
    const float* x  = (const float*)d_in[0];   // (64,32,3,64,64)
    const float* W  = (const float*)d_in[1];   // (12288,24)
    const float* be = (const float*)d_in[2];   // (24,)
    float* out = (float*)d_out;                // (64,31,3,4)
    float* ws  = (float*)d_ws;

    float* part = ws;                             // 128*8*256   = 262144 floats
    float* Bt   = part + (size_t)MTILES * KSPLIT * 256;
    float* z    = Bt + (size_t)16 * KDIM;         // 16*12288    = 196608 floats
                                                  // z: 2048*12  =  24576 floats

    prep_b_kernel<<<(16 * KDIM) / 256, 256, 0, stream>>>(W, Bt);

    dim3 g(MTILES, KSPLIT);
    gemm_wmma_kernel<<<g, 32, 0, stream>>>(x, Bt, part);

    reduce_constrain_kernel<<<(M_ROWS * 12) / 256, 256, 0, stream>>>(part, be, z);

    match_fix_out_kernel<<<NB, 32, 0, stream>>>(z, out);
}